// GridPull_71554155151976
// MI455X (gfx1250) — compile-verified
//
#include <hip/hip_runtime.h>

// grid_pull, trilinear, bound='zero', extrapolate=False
// input: (B=2, C=2, 128,128,128) f32 ; grid: (B=2, 3, 128,128,128) f32 voxel coords
// out:   (B=2, C=2, 128,128,128) f32

constexpr int B  = 2;
constexpr int C  = 2;
constexpr int N  = 128;
constexpr int N3 = N * N * N;            // 2^21
constexpr int LOG_N3 = 21;

// One z-adjacent tap pair, shared helper.
// zhi==true means iz0==N-1 (frac wz is exactly 0): the dz=0 value lives at idx+1
// of the clamped pair and the dz=1 term is multiplied by wz==0.
__device__ __forceinline__ float zpair(const float* __restrict__ p, int idx,
                                       bool zhi, float wz0, float wz1) {
  float lo = p[idx];
  float hi = p[idx + 1];
  float v0 = zhi ? hi : lo;
  return __builtin_fmaf(v0, wz0, hi * wz1);
}

__global__ __launch_bounds__(256) void grid_pull_kernel(
    const float* __restrict__ input,   // (B,C,N,N,N)
    const float* __restrict__ grid,    // (B,3,N,N,N)
    float* __restrict__ out)           // (B,C,N,N,N)
{
  int s = blockIdx.x * blockDim.x + threadIdx.x;   // 0 .. B*N3-1
  int b = s >> LOG_N3;
  int v = s & (N3 - 1);

  const float* gb = grid + (size_t)b * 3 * N3 + v;

  // gfx1250 global_prefetch_b8: pull the three grid streams ahead of the
  // gather-dominated body (streamed exactly once; latency hiding only).
  __builtin_prefetch(gb + 0 * N3 + 4096, 0, 1);
  __builtin_prefetch(gb + 1 * N3 + 4096, 0, 1);
  __builtin_prefetch(gb + 2 * N3 + 4096, 0, 1);

  // Non-temporal: grid is read exactly once; don't pollute L2 (input lives there).
  float gx = __builtin_nontemporal_load(gb + 0 * N3);
  float gy = __builtin_nontemporal_load(gb + 1 * N3);
  float gz = __builtin_nontemporal_load(gb + 2 * N3);

  float* ob = out + (size_t)b * C * N3 + v;

  const float hi = (float)(N - 1);
  bool inb = (gx >= 0.f) & (gx <= hi) &
             (gy >= 0.f) & (gy <= hi) &
             (gz >= 0.f) & (gz <= hi);

  if (!inb) {
    // extrapolate=False: zero outside the field of view; skip all gathers.
    __builtin_nontemporal_store(0.f, ob);
    __builtin_nontemporal_store(0.f, ob + N3);
    return;
  }

  float fx = floorf(gx), fy = floorf(gy), fz = floorf(gz);
  float wx = gx - fx,    wy = gy - fy,    wz = gz - fz;
  int ix0 = (int)fx, iy0 = (int)fy, iz0 = (int)fz;

  // In-FOV => every out-of-range tap has an exactly-zero weight, so clamped
  // indices reproduce the zero-bound reference without per-tap masks.
  int ix1 = min(ix0 + 1, N - 1);
  int iy1 = min(iy0 + 1, N - 1);
  int izl = min(iz0, N - 2);          // z pair (izl, izl+1) always in range
  bool zhi = (iz0 == N - 1);          // then wz == 0 exactly

  float wx0 = 1.f - wx, wy0 = 1.f - wy, wz0 = 1.f - wz;
  float w00 = wx0 * wy0;
  float w01 = wx0 * wy;
  float w10 = wx  * wy0;
  float w11 = wx  * wy;

  int c00 = (ix0 << 14) + (iy0 << 7) + izl;
  int c01 = (ix0 << 14) + (iy1 << 7) + izl;
  int c10 = (ix1 << 14) + (iy0 << 7) + izl;
  int c11 = (ix1 << 14) + (iy1 << 7) + izl;

  const float* p0 = input + (size_t)(b * C + 0) * N3;  // channel 0 slab (8 MB, L2-resident)
  const float* p1 = p0 + N3;                           // channel 1 slab

  // Channel 0: 4 corners x z-pair lerp
  float a0;
  a0 = w00 * zpair(p0, c00, zhi, wz0, wz);
  a0 = __builtin_fmaf(w01, zpair(p0, c01, zhi, wz0, wz), a0);
  a0 = __builtin_fmaf(w10, zpair(p0, c10, zhi, wz0, wz), a0);
  a0 = __builtin_fmaf(w11, zpair(p0, c11, zhi, wz0, wz), a0);

  // Channel 1: identical indices/weights, different slab
  float a1;
  a1 = w00 * zpair(p1, c00, zhi, wz0, wz);
  a1 = __builtin_fmaf(w01, zpair(p1, c01, zhi, wz0, wz), a1);
  a1 = __builtin_fmaf(w10, zpair(p1, c10, zhi, wz0, wz), a1);
  a1 = __builtin_fmaf(w11, zpair(p1, c11, zhi, wz0, wz), a1);

  // Output written exactly once: non-temporal so it streams past L2.
  __builtin_nontemporal_store(a0, ob);
  __builtin_nontemporal_store(a1, ob + N3);
}

extern "C" void kernel_launch(void* const* d_in, const int* in_sizes, int n_in,
                              void* d_out, int out_size, void* d_ws, size_t ws_size,
                              hipStream_t stream) {
  const float* input = (const float*)d_in[0];   // B*C*N3 = 8388608 floats
  const float* grid  = (const float*)d_in[1];   // B*3*N3 = 12582912 floats
  float* out = (float*)d_out;                   // B*C*N3 floats

  const int total  = B * N3;                    // 4,194,304 threads, one per grid point
  const int block  = 256;                       // 8 wave32 waves per block
  const int blocks = (total + block - 1) / block;
  grid_pull_kernel<<<blocks, block, 0, stream>>>(input, grid, out);
}